// GaussianRenderer_47545287966879
// MI455X (gfx1250) — compile-verified
//
#include <hip/hip_runtime.h>

#define N_G   512
#define H_IMG 256
#define W_IMG 256

// ws / LDS layout (floats):
//   [0*512..6*512)   : SoA mx, my, ia, ibc, idd, coef
//   [6*512..10*512)  : colors as float4 per gaussian (r,g,b,0)
#define GC_BASE (6 * N_G)
#define WS_FLOATS (10 * N_G)

typedef float v2f __attribute__((ext_vector_type(2)));
typedef float v8f __attribute__((ext_vector_type(8)));

// ---------------- Kernel 1: project + cov2D inverse + depth bitonic sort ----
__global__ __launch_bounds__(N_G) void gs_preprocess_sort(
    const float* __restrict__ means3D, const float* __restrict__ covs3d,
    const float* __restrict__ colors,  const float* __restrict__ opac,
    const float* __restrict__ Km, const float* __restrict__ Rm,
    const float* __restrict__ tv, float* __restrict__ ws)
{
    __shared__ float P[WS_FLOATS];
    __shared__ float sk[N_G];
    __shared__ int   sv[N_G];
    const int i = threadIdx.x;

    const float R00=Rm[0],R01=Rm[1],R02=Rm[2],
                R10=Rm[3],R11=Rm[4],R12=Rm[5],
                R20=Rm[6],R21=Rm[7],R22=Rm[8];
    const float K00=Km[0],K01=Km[1],K02=Km[2],
                K10=Km[3],K11=Km[4],K12=Km[5],
                K20=Km[6],K21=Km[7],K22=Km[8];
    const float t0=tv[0],t1=tv[1],t2=tv[2];

    const float mx = means3D[i*3+0], my = means3D[i*3+1], mz = means3D[i*3+2];
    const float X = R00*mx + R01*my + R02*mz + t0;
    const float Y = R10*mx + R11*my + R12*mz + t1;
    const float Z = R20*mx + R21*my + R22*mz + t2;
    const float depth = fmaxf(Z, 1.0f);

    const float sx = K00*X + K01*Y + K02*Z;
    const float sy = K10*X + K11*Y + K12*Z;
    const float sz = K20*X + K21*Y + K22*Z;
    const float m2x = sx / sz, m2y = sy / sz;

    const float fx = K00, fy = K11;
    const float iz = 1.0f / Z, iz2 = iz * iz;
    const float J00 = fx*iz,  J02 = -fx*X*iz2;   // J01 = 0
    const float J11 = fy*iz,  J12 = -fy*Y*iz2;   // J10 = 0

    const float* C = covs3d + i*9;
    const float c00=C[0],c01=C[1],c02=C[2],
                c10=C[3],c11=C[4],c12=C[5],
                c20=C[6],c21=C[7],c22=C[8];
    // M = R * C
    const float M00=R00*c00+R01*c10+R02*c20, M01=R00*c01+R01*c11+R02*c21, M02=R00*c02+R01*c12+R02*c22;
    const float M10=R10*c00+R11*c10+R12*c20, M11=R10*c01+R11*c11+R12*c21, M12=R10*c02+R11*c12+R12*c22;
    const float M20=R20*c00+R21*c10+R22*c20, M21=R20*c01+R21*c11+R22*c21, M22=R20*c02+R21*c12+R22*c22;
    // cc = M * R^T
    const float cc00=M00*R00+M01*R01+M02*R02, cc01=M00*R10+M01*R11+M02*R12, cc02=M00*R20+M01*R21+M02*R22;
    const float cc10=M10*R00+M11*R01+M12*R02, cc11=M10*R10+M11*R11+M12*R12, cc12=M10*R20+M11*R21+M12*R22;
    const float cc20=M20*R00+M21*R01+M22*R02, cc21=M20*R10+M21*R11+M22*R12, cc22=M20*R20+M21*R21+M22*R22;
    // V = J * cc   (J01 = J10 = 0)
    const float V00=J00*cc00+J02*cc20, V01=J00*cc01+J02*cc21, V02=J00*cc02+J02*cc22;
    const float V10=J11*cc10+J12*cc20, V11=J11*cc11+J12*cc21, V12=J11*cc12+J12*cc22;
    // cov2D = V * J^T + 1e-4 I
    const float a  = V00*J00 + V02*J02 + 1e-4f;
    const float b  = V01*J11 + V02*J12;
    const float c_ = V10*J00 + V12*J02;
    const float d  = V11*J11 + V12*J12 + 1e-4f;

    const float det = a*d - b*c_;
    const float inv = 1.0f / det;
    const float ia  =  d * inv;
    const float ibc = -(b + c_) * inv;
    const float idd =  a * inv;

    const bool  valid = (depth > 1.0f) && (depth < 50.0f);
    const float norm  = 0.15915494309189535f / sqrtf(det);   // 1/(2*pi*sqrt(det))
    const float coef  = valid ? (opac[i] * norm) : 0.0f;

    P[0*N_G+i] = m2x;  P[1*N_G+i] = m2y;
    P[2*N_G+i] = ia;   P[3*N_G+i] = ibc;  P[4*N_G+i] = idd;
    P[5*N_G+i] = coef;
    P[GC_BASE + i*4 + 0] = colors[i*3+0];
    P[GC_BASE + i*4 + 1] = colors[i*3+1];
    P[GC_BASE + i*4 + 2] = colors[i*3+2];
    P[GC_BASE + i*4 + 3] = 0.0f;                 // zero column for WMMA padding
    sk[i] = depth;  sv[i] = i;
    __syncthreads();

    // bitonic sort ascending by depth (payload = original index)
    for (int k = 2; k <= N_G; k <<= 1) {
        for (int j = k >> 1; j > 0; j >>= 1) {
            const int ixj = i ^ j;
            if (ixj > i) {
                const bool up = ((i & k) == 0);
                const float ki = sk[i], kj = sk[ixj];
                if ((ki > kj) == up) {
                    sk[i] = kj; sk[ixj] = ki;
                    const int v = sv[i]; sv[i] = sv[ixj]; sv[ixj] = v;
                }
            }
            __syncthreads();
        }
    }

    const int src = sv[i];
    #pragma unroll
    for (int f = 0; f < 6; ++f)
        ws[f*N_G + i] = P[f*N_G + src];
    #pragma unroll
    for (int c2 = 0; c2 < 4; ++c2)
        ws[GC_BASE + i*4 + c2] = P[GC_BASE + src*4 + c2];
}

// ---------------- Kernel 2: fused alpha-composite render, WMMA color blend --
// One wave = 16 pixels. A(16x4) = per-chunk composited weights,
// B(4x16)  = gaussian colors replicated mod-4 (col 3,7,11,15 = 0),
// C(16x16) f32 accumulator; only columns 0..2 are stored out.
__global__ __launch_bounds__(256) void gs_render(
    const float* __restrict__ ws, float* __restrict__ out)
{
    __shared__ float g[WS_FLOATS];
    const int tid = threadIdx.x;
    // vectorized LDS staging: 10*512 floats = 1280 float4
    {
        const float4* src4 = (const float4*)ws;
        float4* dst4 = (float4*)g;
        #pragma unroll
        for (int idx = tid; idx < WS_FLOATS / 4; idx += 256) dst4[idx] = src4[idx];
    }
    __syncthreads();

    const int lane = tid & 31;
    const int wave = tid >> 5;
    const int half = lane >> 4;        // 0: K0/K1 slots, 1: K2/K3 slots
    const int m    = lane & 15;        // A-matrix row (pixel), B-matrix col
    const int cm   = m & 3;            // color component (3 -> zero slot)
    const int tileBase = (blockIdx.x * 8 + wave) * 16;

    const int pm = tileBase + m;
    const float px = (float)(pm % W_IMG);
    const float py = (float)(pm / W_IMG);

    v8f acc = {};
    float T = 1.0f;
    const int kbase = half * 2;

    #pragma unroll 4
    for (int chunk = 0; chunk < N_G / 4; ++chunk) {
        const int g0 = chunk * 4 + kbase;
        const int g1 = g0 + 1;

        // two alphas for this lane's K slots at its pixel
        float dx0 = px - g[0*N_G + g0];
        float dx1 = py - g[1*N_G + g0];
        float e0 = -0.5f * (g[2*N_G+g0]*dx0*dx0 + g[3*N_G+g0]*dx0*dx1 + g[4*N_G+g0]*dx1*dx1);
        float al0 = g[5*N_G + g0] * __expf(e0);

        dx0 = px - g[0*N_G + g1];
        dx1 = py - g[1*N_G + g1];
        float e1 = -0.5f * (g[2*N_G+g1]*dx0*dx0 + g[3*N_G+g1]*dx0*dx1 + g[4*N_G+g1]*dx1*dx1);
        float al1 = g[5*N_G + g1] * __expf(e1);

        // exchange alphas with the other half-wave (same pixel, other K slots)
        const float o0 = __shfl_xor(al0, 16, 32);
        const float o1 = __shfl_xor(al1, 16, 32);

        const float aA = half ? o0 : al0;   // alpha at k=0
        const float aB = half ? o1 : al1;   // alpha at k=1
        const float aC = half ? al0 : o0;   // alpha at k=2
        const float aD = half ? al1 : o1;   // alpha at k=3

        // sequential front-to-back transmittance (identical in both halves)
        const float wA = T * aA;
        const float T1 = T  * (1.0f - aA);
        const float wB = T1 * aB;
        const float T2 = T1 * (1.0f - aB);
        const float wC = T2 * aC;
        const float T3 = T2 * (1.0f - aC);
        const float wD = T3 * aD;
        T = T3 * (1.0f - aD);

        v2f A;
        A[0] = half ? wC : wA;             // VGPR0: K = 2*half
        A[1] = half ? wD : wB;             // VGPR1: K = 2*half + 1
        v2f B;                              // unconditional LDS loads
        B[0] = g[GC_BASE + g0*4 + cm];
        B[1] = g[GC_BASE + g1*4 + cm];

        // D = A(16x4) * B(4x16) + C ; full-wave EXEC, f32, exact accumulate
        acc = __builtin_amdgcn_wmma_f32_16x16x4_f32(
            false, A, false, B, (short)0, acc, false, false);
    }

    // C/D layout: VGPR r, lanes 0-15 -> pixel row r, lanes 16-31 -> row r+8;
    // column index = m. Only color columns 0..2 are meaningful.
    if (m < 3) {
        #pragma unroll
        for (int r = 0; r < 8; ++r) {
            const int pix = tileBase + r + 8 * half;
            out[pix * 3 + m] = acc[r];
        }
    }
}

extern "C" void kernel_launch(void* const* d_in, const int* in_sizes, int n_in,
                              void* d_out, int out_size, void* d_ws, size_t ws_size,
                              hipStream_t stream) {
    const float* means3D = (const float*)d_in[0];
    const float* covs3d  = (const float*)d_in[1];
    const float* colors  = (const float*)d_in[2];
    const float* opac    = (const float*)d_in[3];
    const float* Km      = (const float*)d_in[4];
    const float* Rm      = (const float*)d_in[5];
    const float* tv      = (const float*)d_in[6];
    // d_in[7] (pixel meshgrid) is implied by pixel index; not needed.

    float* ws  = (float*)d_ws;    // 10*512 floats, depth-sorted SoA + color4
    float* out = (float*)d_out;   // H*W*3 f32

    gs_preprocess_sort<<<1, N_G, 0, stream>>>(means3D, covs3d, colors, opac,
                                              Km, Rm, tv, ws);

    // 65536 pixels / (16 pix/wave * 8 waves/block) = 512 blocks
    gs_render<<<(H_IMG * W_IMG) / 128, 256, 0, stream>>>(ws, out);
}